// ContractingREN_72138270704171
// MI455X (gfx1250) — compile-verified
//
#include <hip/hip_runtime.h>
#include <math.h>

// Problem constants (from reference)
#define NX 256
#define NY 128
#define NU 128
#define NQ 256
#define NB 4096
#define NTOT 768          // 2*NX + NQ
static constexpr float ALPHA_C   = 1.0f;
static constexpr float EPSILON_C = 0.001f;

typedef __attribute__((ext_vector_type(2))) float v2f;
typedef __attribute__((ext_vector_type(8))) float v8f;

// ---------------------------------------------------------------------------
// fp32 WMMA GEMM, 32x64 tile per wave (2x4 register blocking):
//   C[MxN] = accum*C + alpha * A[MxK] @ B[NxK]^T
// Row-major A and B; B used transposed (C[m,n] = sum_k A[m,k]*B[n,k]).
//
// V_WMMA_F32_16X16X4_F32 operand layouts (CDNA5 ISA, wave32):
//   A 16x4 : lane = (k>>1)*16 + m, vgpr = k&1  -> lane m loads float2 A[m][kk+2*half]
//   B 4x16 : lane = (k>>1)*16 + n, vgpr = k&1  -> lane n loads float2 B[n][kk+2*half]
//   C 16x16: vgpr r -> (m = r + 8*half, n = lane&15)
// Per K=4 step: 6 float2 loads feed 8 WMMAs (A reused 4x, B reused 2x).
// Requires M % 32 == 0, N % 64 == 0, K % 4 == 0 (true for all call sites).
// ---------------------------------------------------------------------------
__global__ void wmma_gemm_abt(const float* __restrict__ A, int lda,
                              const float* __restrict__ B, int ldb,
                              float* __restrict__ C, int ldc,
                              int M, int N, int K,
                              float alpha, int accumulate)
{
    const int wavesPerBlock = blockDim.x >> 5;
    const int wave   = blockIdx.x * wavesPerBlock + (threadIdx.x >> 5);
    const int tilesN = N >> 6;                    // 64-wide N tiles
    const int tilesM = M >> 5;                    // 32-tall M tiles
    if (wave >= tilesM * tilesN) return;          // wave-uniform exit

    const int tm   = wave / tilesN;
    const int tn   = wave % tilesN;
    const int lane = threadIdx.x & 31;
    const int half = lane >> 4;                   // 0: K pair {0,1}, 1: K pair {2,3}
    const int idx  = lane & 15;                   // M index for A, N index for B

    const float* __restrict__ Arow0 = A + (size_t)(tm * 32 + idx)      * lda + 2 * half;
    const float* __restrict__ Arow1 = A + (size_t)(tm * 32 + 16 + idx) * lda + 2 * half;
    const float* __restrict__ Brow0 = B + (size_t)(tn * 64 + idx)      * ldb + 2 * half;
    const float* __restrict__ Brow1 = B + (size_t)(tn * 64 + 16 + idx) * ldb + 2 * half;
    const float* __restrict__ Brow2 = B + (size_t)(tn * 64 + 32 + idx) * ldb + 2 * half;
    const float* __restrict__ Brow3 = B + (size_t)(tn * 64 + 48 + idx) * ldb + 2 * half;

    v8f acc00 = {}, acc01 = {}, acc02 = {}, acc03 = {};
    v8f acc10 = {}, acc11 = {}, acc12 = {}, acc13 = {};

    for (int k = 0; k < K; k += 4) {
        v2f a0 = *(const v2f*)(Arow0 + k);
        v2f a1 = *(const v2f*)(Arow1 + k);
        v2f b0 = *(const v2f*)(Brow0 + k);
        v2f b1 = *(const v2f*)(Brow1 + k);
        v2f b2 = *(const v2f*)(Brow2 + k);
        v2f b3 = *(const v2f*)(Brow3 + k);
        acc00 = __builtin_amdgcn_wmma_f32_16x16x4_f32(false, a0, false, b0, (short)0, acc00, false, false);
        acc01 = __builtin_amdgcn_wmma_f32_16x16x4_f32(false, a0, false, b1, (short)0, acc01, false, false);
        acc02 = __builtin_amdgcn_wmma_f32_16x16x4_f32(false, a0, false, b2, (short)0, acc02, false, false);
        acc03 = __builtin_amdgcn_wmma_f32_16x16x4_f32(false, a0, false, b3, (short)0, acc03, false, false);
        acc10 = __builtin_amdgcn_wmma_f32_16x16x4_f32(false, a1, false, b0, (short)0, acc10, false, false);
        acc11 = __builtin_amdgcn_wmma_f32_16x16x4_f32(false, a1, false, b1, (short)0, acc11, false, false);
        acc12 = __builtin_amdgcn_wmma_f32_16x16x4_f32(false, a1, false, b2, (short)0, acc12, false, false);
        acc13 = __builtin_amdgcn_wmma_f32_16x16x4_f32(false, a1, false, b3, (short)0, acc13, false, false);
    }

    const v8f* accs[2][4] = {{&acc00, &acc01, &acc02, &acc03},
                             {&acc10, &acc11, &acc12, &acc13}};
#pragma unroll
    for (int mi = 0; mi < 2; ++mi) {
#pragma unroll
        for (int nj = 0; nj < 4; ++nj) {
            float* __restrict__ Crow =
                C + (size_t)(tm * 32 + mi * 16 + 8 * half) * ldc + tn * 64 + nj * 16 + idx;
            const v8f acc = *accs[mi][nj];
#pragma unroll
            for (int r = 0; r < 8; ++r) {
                float v = alpha * acc[r];
                if (accumulate) v += Crow[(size_t)r * ldc];
                Crow[(size_t)r * ldc] = v;
            }
        }
    }
}

// ---------------------------------------------------------------------------
// Prep: from full H = X@X^T (no eps) and Y, build
//   E   = 0.5*(H11 + H33/a^2 + Y - Y^T)  (+ eps*(1+1/a^2)/2 on diag)
//   lam = 0.5*(diag(H22) + eps)
//   D11 = -strict_lower(H22)
// ---------------------------------------------------------------------------
__global__ void prep_kernel(const float* __restrict__ H, const float* __restrict__ Y,
                            float* __restrict__ E, float* __restrict__ lam,
                            float* __restrict__ D11)
{
    const int i = blockIdx.x;
    const int j = threadIdx.x;
    const float inva2 = 1.0f / (ALPHA_C * ALPHA_C);

    float h11 = H[(size_t)i * NTOT + j];
    float h33 = H[(size_t)(2 * NX + i) * NTOT + (2 * NX + j)];
    float e = 0.5f * (h11 + h33 * inva2 +
                      Y[(size_t)i * NX + j] - Y[(size_t)j * NX + i]);
    if (i == j) e += 0.5f * (EPSILON_C + EPSILON_C * inva2);
    E[(size_t)i * NX + j] = e;

    float h22 = H[(size_t)(NX + i) * NTOT + (NX + j)];
    D11[(size_t)i * NQ + j] = (j < i) ? -h22 : 0.0f;
    if (i == j) lam[i] = 0.5f * (h22 + EPSILON_C);
}

// ---------------------------------------------------------------------------
// 256x256 Gauss-Jordan inversion, single workgroup of 256 threads.
// Maug is a 256x512 scratch [E | I]; pivot row cached in LDS.
// ---------------------------------------------------------------------------
__global__ void invert256_kernel(const float* __restrict__ E,
                                 float* __restrict__ Maug,
                                 float* __restrict__ Einv)
{
    const int t = threadIdx.x;                 // 0..255
    __shared__ float piv[512];

    for (int r = 0; r < NX; ++r) {
        Maug[(size_t)r * 512 + t]       = E[(size_t)r * NX + t];
        Maug[(size_t)r * 512 + 256 + t] = (r == t) ? 1.0f : 0.0f;
    }
    __syncthreads();

    for (int k = 0; k < NX; ++k) {
        float p = Maug[(size_t)k * 512 + k];
        __syncthreads();
        float inv = 1.0f / p;
        Maug[(size_t)k * 512 + t]       *= inv;
        Maug[(size_t)k * 512 + 256 + t] *= inv;
        __syncthreads();
        piv[t]       = Maug[(size_t)k * 512 + t];
        piv[256 + t] = Maug[(size_t)k * 512 + 256 + t];
        __syncthreads();
        if (t != k) {
            float f = Maug[(size_t)t * 512 + k];
            for (int c = 0; c < 512; ++c)
                Maug[(size_t)t * 512 + c] -= f * piv[c];
        }
        __syncthreads();
    }

    for (int r = 0; r < NX; ++r)
        Einv[(size_t)r * NX + t] = Maug[(size_t)r * 512 + 256 + t];
}

// ---------------------------------------------------------------------------
// Sequential part of the tanh recurrence for one block of 16 columns.
// acc holds base plus cross-block contributions; bv added here exactly once.
// One thread per batch row.
// ---------------------------------------------------------------------------
__global__ void scan_block_kernel(const float* __restrict__ D11,
                                  const float* __restrict__ lam,
                                  const float* __restrict__ bv,
                                  const float* __restrict__ acc,
                                  float* __restrict__ w,
                                  int kb)
{
    const int b  = blockIdx.x * blockDim.x + threadIdx.x;   // 0..NB-1
    const int k0 = kb * 16;

    float a[16], wl[16];
#pragma unroll
    for (int i = 0; i < 16; ++i)
        a[i] = acc[(size_t)b * NQ + k0 + i] + bv[k0 + i];

#pragma unroll
    for (int i = 0; i < 16; ++i) {
        const int k = k0 + i;
        float v = a[i];
        for (int j = 0; j < i; ++j)
            v += wl[j] * D11[(size_t)k * NQ + k0 + j];
        v /= lam[k];
        wl[i] = tanhf(v);
    }

#pragma unroll
    for (int i = 0; i < 16; ++i)
        w[(size_t)b * NQ + k0 + i] = wl[i];
}

// ---------------------------------------------------------------------------
static inline void launch_gemm(const float* A, int lda, const float* B, int ldb,
                               float* C, int ldc, int M, int N, int K,
                               float alpha, int accumulate, hipStream_t stream)
{
    const int tiles  = (M / 32) * (N / 64);
    const int wavesB = 8;                         // 256 threads / wave32
    const int blocks = (tiles + wavesB - 1) / wavesB;
    hipLaunchKernelGGL(wmma_gemm_abt, dim3(blocks), dim3(256), 0, stream,
                       A, lda, B, ldb, C, ldc, M, N, K, alpha, accumulate);
}

extern "C" void kernel_launch(void* const* d_in, const int* in_sizes, int n_in,
                              void* d_out, int out_size, void* d_ws, size_t ws_size,
                              hipStream_t stream)
{
    (void)in_sizes; (void)n_in; (void)out_size; (void)ws_size;

    const float* x   = (const float*)d_in[0];   // NB x NX
    const float* u   = (const float*)d_in[1];   // NB x NU
    const float* X   = (const float*)d_in[2];   // NTOT x NTOT
    const float* Y   = (const float*)d_in[3];   // NX x NX
    const float* D12 = (const float*)d_in[4];   // NQ x NU
    const float* B2  = (const float*)d_in[5];   // NX x NU
    const float* C2  = (const float*)d_in[6];   // NY x NX
    const float* D21 = (const float*)d_in[7];   // NY x NQ
    const float* D22 = (const float*)d_in[8];   // NY x NU
    const float* bv  = (const float*)d_in[9];   // NQ

    float* dx = (float*)d_out;                  // NB x NX
    float* y  = dx + (size_t)NB * NX;           // NB x NY

    // workspace carve-up (floats)
    float* ws   = (float*)d_ws;
    float* H    = ws;                      ws += (size_t)NTOT * NTOT;   // 768x768
    float* E    = ws;                      ws += (size_t)NX * NX;
    float* Maug = ws;                      ws += (size_t)NX * 512;
    float* Einv = ws;                      ws += (size_t)NX * NX;
    float* lam  = ws;                      ws += NQ;
    float* D11  = ws;                      ws += (size_t)NQ * NQ;
    float* acc  = ws;                      ws += (size_t)NB * NQ;       // base + updates
    float* w    = ws;                      ws += (size_t)NB * NQ;
    float* s    = ws;                      ws += (size_t)NB * NX;

    // 1) H = X @ X^T   (eps handled in prep)
    launch_gemm(X, NTOT, X, NTOT, H, NTOT, NTOT, NTOT, NTOT, 1.0f, 0, stream);

    // 2) E, lam, D11 from H blocks + Y
    hipLaunchKernelGGL(prep_kernel, dim3(NX), dim3(NX), 0, stream, H, Y, E, lam, D11);

    // 3) Einv = inv(E)
    hipLaunchKernelGGL(invert256_kernel, dim3(1), dim3(256), 0, stream, E, Maug, Einv);

    // 4) acc = x @ C1^T + u @ D12^T    (C1 = -H21  ->  alpha = -1 with B = H21)
    const float* H21 = H + (size_t)NX * NTOT;            // rows 256.., cols 0..
    launch_gemm(x, NX, H21, NTOT, acc, NQ, NB, NQ, NX, -1.0f, 0, stream);
    launch_gemm(u, NU, D12, NU,   acc, NQ, NB, NQ, NU,  1.0f, 1, stream);

    // 5) blocked tanh recurrence: 16 sequential columns per step + WMMA panel
    //    update.  The update writes ALL 256 columns: for already-consumed
    //    columns the D11 structure gives zeros or the column is never read
    //    again, so full-width update keeps N a multiple of 64.
    for (int kb = 0; kb < NQ / 16; ++kb) {
        hipLaunchKernelGGL(scan_block_kernel, dim3(NB / 256), dim3(256), 0, stream,
                           D11, lam, bv, acc, w, kb);
        if (kb < NQ / 16 - 1) {
            const int k0 = kb * 16;
            // acc[:, :] += w[:, k0:k0+16] @ D11[:, k0:k0+16]^T
            launch_gemm(w + k0, NQ,
                        D11 + k0, NQ,
                        acc, NQ,
                        NB, NQ, 16, 1.0f, 1, stream);
        }
    }

    // 6) s = x @ F^T + w @ B1^T + u @ B2^T    (F = H31, B1 = H32)
    const float* H31 = H + (size_t)(2 * NX) * NTOT;          // rows 512.., cols 0..
    const float* H32 = H + (size_t)(2 * NX) * NTOT + NX;     // rows 512.., cols 256..
    launch_gemm(x, NX, H31, NTOT, s, NX, NB, NX, NX, 1.0f, 0, stream);
    launch_gemm(w, NQ, H32, NTOT, s, NX, NB, NX, NQ, 1.0f, 1, stream);
    launch_gemm(u, NU, B2,  NU,   s, NX, NB, NX, NU, 1.0f, 1, stream);

    // 7) dx = s @ Einv^T
    launch_gemm(s, NX, Einv, NX, dx, NX, NB, NX, NX, 1.0f, 0, stream);

    // 8) y = x @ C2^T + w @ D21^T + u @ D22^T
    launch_gemm(x, NX, C2,  NX, y, NY, NB, NY, NX, 1.0f, 0, stream);
    launch_gemm(w, NQ, D21, NQ, y, NY, NB, NY, NQ, 1.0f, 1, stream);
    launch_gemm(u, NU, D22, NU, y, NY, NB, NY, NU, 1.0f, 1, stream);
}